// CompoundEmbedding_42666205119354
// MI455X (gfx1250) — compile-verified
//
#include <hip/hip_runtime.h>
#include <stdint.h>
#include <stddef.h>

// EmbeddingBag(sum): out[b,:] = sum_{l<BAG} weight[idx[b,l],:]
// B=16384, BAG=50, D=128 (f32). Weight table (~51MB) is L2-resident on MI455X
// (192MB L2) -> L2-bandwidth/latency-bound random gather. Data path: CDNA5
// async global->LDS gather (GLOBAL_LOAD_ASYNC_TO_LDS_B128, saddr/GVS form,
// tracked by ASYNCcnt) with a double-buffered software pipeline; one wave32
// per bag, one async op moves one full 512B embedding row into LDS.
//
//  * all 50 bag indices pulled into SGPRs up-front (wide s_loads, single
//    s_wait_kmcnt) instead of a serialized s_load/s_wait chain per row
//  * async issues use the scalar-base (GVS) addressing form: per-row base in
//    SGPRs (SALU), one shared lane*16 VGPR offset -> no per-row v_add_u64
//  * output uses non-temporal b128 stores (native clang vector type, since
//    the builtin rejects HIP's float4 class) so the streamed 8.4MB result
//    does not evict the L2-resident weight table

typedef __attribute__((ext_vector_type(4))) float v4f;

#define EMB_DIM 128
#define BAG 50
#define WAVES_PER_BLOCK 8
#define BLOCK_THREADS (WAVES_PER_BLOCK * 32)
#define CHUNK 5                       // rows per pipeline stage
#define NCHUNK (BAG / CHUNK)          // 10
#define ROW_BYTES (EMB_DIM * 4)       // 512
#define LDS_PER_WAVE (2 * CHUNK * ROW_BYTES)  // 5120 B (double buffer)

__global__ __launch_bounds__(BLOCK_THREADS)
void embbag_sum_kernel(const int* __restrict__ indices,
                       const float* __restrict__ weight,
                       float* __restrict__ out,
                       int batch) {
    __shared__ __align__(16) unsigned char smem[WAVES_PER_BLOCK * LDS_PER_WAVE];

    const int lane = threadIdx.x & 31;
    // Force wave id into an SGPR so per-bag accesses scalarize.
    const int wid  = __builtin_amdgcn_readfirstlane((int)(threadIdx.x >> 5));
    const int bag  = (int)blockIdx.x * WAVES_PER_BLOCK + wid;
    if (bag >= batch) return;

    const int* __restrict__ ib = indices + (size_t)bag * BAG;

    // Pull the whole bag's indices into SGPRs first: uniform address -> wide
    // scalar loads, and a single kmcnt wait instead of 50 serialized ones.
    uint32_t idxbuf[BAG];
    #pragma unroll
    for (int l = 0; l < BAG; ++l) idxbuf[l] = (uint32_t)ib[l];

    // Low 32 bits of a flat shared-memory address are the LDS byte offset.
    const uint32_t lds_wave = (uint32_t)(uintptr_t)(&smem[0])
                            + (uint32_t)(wid * LDS_PER_WAVE);
    const uint32_t lane16 = (uint32_t)lane * 16u;   // shared vaddr for all rows

    // Issue one async gather: 32 lanes x 16B = one 512B embedding row -> LDS.
    // GVS form: scalar 64-bit row base + 32-bit per-lane VGPR offset.
    auto issue_row = [&](int l, int buf) {
        const float* rowbase = weight + (size_t)idxbuf[l] * EMB_DIM; // SALU
        const uint32_t ldsa = lds_wave
                            + (uint32_t)(buf * (CHUNK * ROW_BYTES))
                            + (uint32_t)((l % CHUNK) * ROW_BYTES)
                            + lane16;
        asm volatile("global_load_async_to_lds_b128 %0, %1, %2"
                     :
                     : "v"(ldsa), "v"(lane16), "s"(rowbase)
                     : "memory");
    };

    float a0 = 0.f, a1 = 0.f, a2 = 0.f, a3 = 0.f;

    // Prologue: fill buffer 0.
    #pragma unroll
    for (int l = 0; l < CHUNK; ++l) issue_row(l, 0);

    #pragma unroll
    for (int ch = 0; ch < NCHUNK; ++ch) {
        if (ch + 1 < NCHUNK) {
            // Issue next chunk into the other buffer, then wait for the
            // current chunk. Async loads complete in order, so asynccnt<=CHUNK
            // guarantees the oldest CHUNK ops (chunk `ch`) have landed in LDS.
            #pragma unroll
            for (int l = (ch + 1) * CHUNK; l < (ch + 2) * CHUNK; ++l)
                issue_row(l, (ch + 1) & 1);
            asm volatile("s_wait_asynccnt 5" ::: "memory");
        } else {
            asm volatile("s_wait_asynccnt 0" ::: "memory");
        }

        const unsigned char* buf = smem
            + (uint32_t)(wid * LDS_PER_WAVE)
            + (uint32_t)((ch & 1) * (CHUNK * ROW_BYTES));
        #pragma unroll
        for (int r = 0; r < CHUNK; ++r) {
            const v4f v = *(const v4f*)(buf + r * ROW_BYTES + lane16);
            a0 += v.x; a1 += v.y; a2 += v.z; a3 += v.w;
        }
    }

    v4f res;
    res.x = a0; res.y = a1; res.z = a2; res.w = a3;
    // Streamed output, never re-read by this kernel: non-temporal store keeps
    // the weight table resident in L2.
    __builtin_nontemporal_store(res, (v4f*)(out + (size_t)bag * EMB_DIM + (lane << 2)));
}

extern "C" void kernel_launch(void* const* d_in, const int* in_sizes, int n_in,
                              void* d_out, int out_size, void* d_ws, size_t ws_size,
                              hipStream_t stream) {
    const int*   indices = (const int*)d_in[0];   // [B, 50] int32
    const float* weight  = (const float*)d_in[1]; // [100001, 128] f32
    float*       out     = (float*)d_out;         // [B, 128] f32

    const int batch  = in_sizes[0] / BAG;
    const int blocks = (batch + WAVES_PER_BLOCK - 1) / WAVES_PER_BLOCK;

    embbag_sum_kernel<<<blocks, BLOCK_THREADS, 0, stream>>>(indices, weight, out, batch);
}